// LabelSmoothingLossFunction_85478439125743
// MI455X (gfx1250) — compile-verified
//
#include <hip/hip_runtime.h>
#include <cmath>

typedef __attribute__((ext_vector_type(2))) float v2f;
typedef __attribute__((ext_vector_type(4))) float f32x4;
typedef __attribute__((ext_vector_type(8))) float v8f;

#define WAVES_PER_BLOCK 8

// One wave per row: stream 32000 f32 with coalesced nontemporal b128 loads,
// reduce lanes with V_WMMA_F32_16X16X4_F32 (B = ones), apply the closed-form
// label-smoothing row term, serially sum the 8 wave terms per block into d_ws.
__global__ __launch_bounds__(256) void ls_rowsum_kernel(
    const float* __restrict__ yhat, const int* __restrict__ target,
    float* __restrict__ partials, int n_rows, int vocab,
    float base, float conf_minus_base, float crow)
{
    __shared__ float wsums[WAVES_PER_BLOCK];
    const int wave = threadIdx.x >> 5;
    const int lane = threadIdx.x & 31;
    const int row  = blockIdx.x * WAVES_PER_BLOCK + wave;   // wave-uniform

    float term = 0.0f;
    if (row < n_rows) {                                     // uniform branch: EXEC stays all-1s
        const float* rp = yhat + (size_t)row * (size_t)vocab;

        f32x4 acc0 = {0.f, 0.f, 0.f, 0.f};
        f32x4 acc1 = {0.f, 0.f, 0.f, 0.f};
        int i = lane * 4;
        const int iters = vocab / 256;                      // 32000/256 = 125, exact
        for (int it = 0; it < iters; ++it) {
            // prefetch ~4KB ahead (global_prefetch_b8), harmless past the end of the row
            __builtin_prefetch((const void*)(rp + i + 1024), 0, 0);
            f32x4 a = __builtin_nontemporal_load((const f32x4*)(rp + i));
            f32x4 b = __builtin_nontemporal_load((const f32x4*)(rp + i + 128));
            acc0 += a;
            acc1 += b;
            i += 256;
        }
        acc0 += acc1;
        float p = (acc0.x + acc0.y) + (acc0.z + acc0.w);    // per-lane partial

        // Wave-wide reduction via WMMA: A[m][0]=p_m (lanes 0-15), A[m][2]=p_{m+16}
        // (lanes 16-31), A[m][1]=A[m][3]=0, B = ones 4x16.
        // => D[m][n] = p_m + p_{m+16} for every n.
        v2f a2; a2.x = p;   a2.y = 0.0f;
        v2f bb; bb.x = 1.0f; bb.y = 1.0f;
        v8f c = {};
        c = __builtin_amdgcn_wmma_f32_16x16x4_f32(
                /*neg_a=*/false, a2, /*neg_b=*/false, bb,
                /*c_mod=*/(short)0, c, /*reuse_a=*/false, /*reuse_b=*/false);
        // lanes 0-15 hold D rows 0..7, lanes 16-31 hold D rows 8..15 (any N column)
        float hsum = ((c[0] + c[1]) + (c[2] + c[3])) + ((c[4] + c[5]) + (c[6] + c[7]));
        float S = hsum + __shfl_xor(hsum, 16, 32);          // full 32-lane sum

        if (lane == 0) {
            int t = target[row];
            if (t != 0) {   // padding rows contribute 0
                float y0 = rp[0];
                float yt = rp[t];
                term = crow - base * S + base * y0 - conf_minus_base * yt;
            }
        }
    }
    if (lane == 0) wsums[wave] = term;
    __syncthreads();
    if (threadIdx.x == 0) {
        float s = 0.0f;
        #pragma unroll
        for (int w = 0; w < WAVES_PER_BLOCK; ++w) s += wsums[w];
        partials[blockIdx.x] = s;
    }
}

// Deterministic fixed-order final reduction of block partials.
__global__ __launch_bounds__(256) void ls_final_reduce(
    const float* __restrict__ partials, int n, float* __restrict__ out)
{
    __shared__ float sm[256];
    float s = 0.0f;
    for (int i = threadIdx.x; i < n; i += 256) s += partials[i];
    sm[threadIdx.x] = s;
    __syncthreads();
    for (int off = 128; off > 0; off >>= 1) {
        if ((int)threadIdx.x < off) sm[threadIdx.x] += sm[threadIdx.x + off];
        __syncthreads();
    }
    if (threadIdx.x == 0) out[0] = sm[0];
}

extern "C" void kernel_launch(void* const* d_in, const int* in_sizes, int n_in,
                              void* d_out, int out_size, void* d_ws, size_t ws_size,
                              hipStream_t stream) {
    const float* yhat   = (const float*)d_in[0];
    const int*   target = (const int*)d_in[1];
    float*       out    = (float*)d_out;

    const int n_rows = in_sizes[1];              // 4096
    const int vocab  = in_sizes[0] / n_rows;     // 32000

    // Closed-form per-row constants (host-side, double precision).
    const double SM     = 0.1;
    const double conf   = 1.0 - SM;
    const double base_d = SM / (double)(vocab - 2);
    const double crow_d = conf * std::log(conf) + SM * std::log(base_d);

    const float base = (float)base_d;
    const float cmb  = (float)(conf - base_d);
    const float crow = (float)crow_d;

    const int blocks = (n_rows + WAVES_PER_BLOCK - 1) / WAVES_PER_BLOCK;  // 512
    float* partials = (float*)d_ws;              // 512 floats << ws_size

    ls_rowsum_kernel<<<blocks, 256, 0, stream>>>(yhat, target, partials,
                                                 n_rows, vocab, base, cmb, crow);
    ls_final_reduce<<<1, 256, 0, stream>>>(partials, blocks, out);
}